// CubicFourierProductBlock64_18124761989262
// MI455X (gfx1250) — compile-verified
//
#include <hip/hip_runtime.h>
#include <math.h>
#include <stdint.h>

typedef __attribute__((ext_vector_type(2))) float v2f;
typedef __attribute__((ext_vector_type(8))) float v8f;

#define U       64
#define BROWS   4096
#define DIN     66
#define NPROD1  2080
#define NPROD2  45760
#define KTOT    (U + NPROD1 + NPROD2)   // 47904
#define CHUNKS  (KTOT / 4)              // 11976
#define NSEG    4
#define SEG     (CHUNKS / NSEG)         // 2994

// ---- ws layout (bytes) ----
#define OFF_F    ((size_t)0)                       // 4096*64*4   = 1048576
#define OFF_IDX  ((size_t)1048576)                 // 47904*4     = 191616 (pad to 192K)
#define OFF_BP   ((size_t)(1048576 + 196608))      // 11976*1024  = 12263424
#define OFF_PART ((size_t)(1048576 + 196608 + 12263424)) // 4*4096*64*4 = 4194304

// Fourier features: f[row][i] = cos(in[row][i]*fw1[i]+fb1[i])*fw2[i]
__global__ __launch_bounds__(256)
void f_kernel(const float* __restrict__ in, const float* __restrict__ fw1,
              const float* __restrict__ fw2, const float* __restrict__ fb1,
              float* __restrict__ fbuf) {
    int t = blockIdx.x * 256 + threadIdx.x;
    if (t >= BROWS * U) return;
    int row = t >> 6, i = t & 63;
    float x = in[row * DIN + i];
    fbuf[t] = cosf(x * fw1[i] + fb1[i]) * fw2[i];
}

// Packed (a,b,c) index per k; index 64 == "multiply by 1.0"
__global__ __launch_bounds__(256)
void build_idx_kernel(uint32_t* __restrict__ idx) {
    int t = blockIdx.x * 256 + threadIdx.x;
    if (t < U) {                               // linear features
        idx[t] = (uint32_t)t | (64u << 8) | (64u << 16);
    }
    int p = t - U;
    if (p >= 0 && p < NPROD1) {
        // decode pair p -> (a,b), b>=a, row-major
        int a = 0, rem = p;
        while (rem >= U - a) { rem -= U - a; ++a; }
        int b = a + rem;
        idx[U + p] = (uint32_t)a | ((uint32_t)b << 8) | (64u << 16);
        // triples for this pair start at closed-form tetrahedral offset
        long Tet64 = (long)U * (U + 1) * (U + 2) / 6;     // 45760
        long na = U - a, nb = U - b;
        long off = (Tet64 - na * (na + 1) * (na + 2) / 6)
                 + (na * (na + 1) / 2 - nb * (nb + 1) / 2);
        uint32_t base = (uint32_t)(U + NPROD1) + (uint32_t)off;
        for (int c = b; c < U; ++c)
            idx[base + (c - b)] = (uint32_t)a | ((uint32_t)b << 8) | ((uint32_t)c << 16);
    }
}

// Repack wo[K][64] into per-chunk WMMA B-fragment order:
// Bp[chunk][lane][j*2+v] = wo[chunk*4 + 2*(lane/16) + v][16*j + lane%16]
__global__ __launch_bounds__(256)
void pack_b_kernel(const float* __restrict__ wo, float* __restrict__ Bp) {
    size_t t = (size_t)blockIdx.x * 256 + threadIdx.x;
    if (t >= (size_t)CHUNKS * 256) return;
    int chunk = (int)(t >> 8);
    int r     = (int)(t & 255);
    int lane  = r >> 3;
    int j     = (r >> 1) & 3;
    int v     = r & 1;
    int k = chunk * 4 + 2 * (lane >> 4) + v;
    int n = 16 * j + (lane & 15);
    Bp[t] = wo[(size_t)k * U + n];
}

// Fused on-the-fly GEMM: 4 waves/block = 4 K-segments over one 16-row tile.
__global__ __launch_bounds__(128)
void gemm_kernel(const float* __restrict__ fbuf,
                 const uint32_t* __restrict__ idx,
                 const float* __restrict__ Bp,
                 float* __restrict__ partials) {
    __shared__ float fl[16 * 65];
    const int m0   = blockIdx.x * 16;
    const int tid  = threadIdx.x;
    const int seg  = tid >> 5;
    const int lane = tid & 31;
    const int half = lane >> 4;
    const int rowl = lane & 15;

    // stage 16 rows of f into LDS (stride 65, slot 64 = identity 1.0)
    for (int i = tid; i < 16 * U; i += 128) {
        int r = i >> 6, c = i & 63;
        fl[r * 65 + c] = fbuf[(size_t)(m0 + r) * U + c];
    }
    if (tid < 16) fl[tid * 65 + 64] = 1.0f;
    __syncthreads();

    const float* frow = &fl[rowl * 65];
    v8f acc0 = {0.f,0.f,0.f,0.f,0.f,0.f,0.f,0.f};
    v8f acc1 = acc0, acc2 = acc0, acc3 = acc0;

    const int c0 = seg * SEG;
    const uint32_t* ip = idx + (size_t)c0 * 4 + half * 2;   // k = 4c + 2*half + {0,1}
    const float*    bp = Bp + (size_t)c0 * 256 + lane * 8;

    #pragma unroll 2
    for (int c = 0; c < SEG; ++c) {
        uint32_t k0 = ip[0], k1 = ip[1];
        // A fragment: comb values = f[a]*f[b]*f[c] (slot 64 => *1.0)
        float a0 = frow[k0 & 255] * frow[(k0 >> 8) & 255] * frow[(k0 >> 16) & 255];
        float a1 = frow[k1 & 255] * frow[(k1 >> 8) & 255] * frow[(k1 >> 16) & 255];
        v2f A; A.x = a0; A.y = a1;
        // B fragments: 8 contiguous floats per lane (2 x b128)
        float4 b01 = *(const float4*)(bp);
        float4 b23 = *(const float4*)(bp + 4);
        v2f B0; B0.x = b01.x; B0.y = b01.y;
        v2f B1; B1.x = b01.z; B1.y = b01.w;
        v2f B2; B2.x = b23.x; B2.y = b23.y;
        v2f B3; B3.x = b23.z; B3.y = b23.w;

        acc0 = __builtin_amdgcn_wmma_f32_16x16x4_f32(false, A, false, B0, (short)0, acc0, false, false);
        acc1 = __builtin_amdgcn_wmma_f32_16x16x4_f32(false, A, false, B1, (short)0, acc1, false, false);
        acc2 = __builtin_amdgcn_wmma_f32_16x16x4_f32(false, A, false, B2, (short)0, acc2, false, false);
        acc3 = __builtin_amdgcn_wmma_f32_16x16x4_f32(false, A, false, B3, (short)0, acc3, false, false);

        ip += 4;
        bp += 256;
    }

    // D layout: vgpr r, lanes0-15 -> M=r, lanes16-31 -> M=8+r; N = lane%16
    float* po = partials + ((size_t)seg * BROWS + m0) * U + rowl;
    #pragma unroll
    for (int r = 0; r < 8; ++r) {
        float* q = po + (size_t)(r + 8 * half) * U;
        q[0]  = acc0[r];
        q[16] = acc1[r];
        q[32] = acc2[r];
        q[48] = acc3[r];
    }
}

// out = inputs + sum of 4 K-segment partials (cols 64..65 pass-through)
__global__ __launch_bounds__(256)
void reduce_kernel(const float* __restrict__ in, const float* __restrict__ partials,
                   float* __restrict__ out) {
    int t = blockIdx.x * 256 + threadIdx.x;
    if (t >= BROWS * DIN) return;
    int row = t / DIN, col = t - row * DIN;
    float v = in[t];
    if (col < U) {
        size_t base = (size_t)row * U + col;
        const size_t S = (size_t)BROWS * U;
        v += partials[base] + partials[S + base] + partials[2 * S + base] + partials[3 * S + base];
    }
    out[t] = v;
}

extern "C" void kernel_launch(void* const* d_in, const int* in_sizes, int n_in,
                              void* d_out, int out_size, void* d_ws, size_t ws_size,
                              hipStream_t stream) {
    const float* inputs = (const float*)d_in[0];
    const float* fw1    = (const float*)d_in[1];
    const float* fw2    = (const float*)d_in[2];
    const float* fb1    = (const float*)d_in[3];
    const float* wo     = (const float*)d_in[4];
    float* out = (float*)d_out;

    char* ws = (char*)d_ws;
    float*    fbuf     = (float*)(ws + OFF_F);
    uint32_t* idx      = (uint32_t*)(ws + OFF_IDX);
    float*    Bp       = (float*)(ws + OFF_BP);
    float*    partials = (float*)(ws + OFF_PART);

    f_kernel<<<(BROWS * U + 255) / 256, 256, 0, stream>>>(inputs, fw1, fw2, fb1, fbuf);
    build_idx_kernel<<<(U + NPROD1 + 255) / 256, 256, 0, stream>>>(idx);
    pack_b_kernel<<<CHUNKS, 256, 0, stream>>>(wo, Bp);
    gemm_kernel<<<BROWS / 16, 128, 0, stream>>>(fbuf, idx, Bp, partials);
    reduce_kernel<<<(BROWS * DIN + 255) / 256, 256, 0, stream>>>(inputs, partials, out);
}